// ExtractorHead_29326036697744
// MI455X (gfx1250) — compile-verified
//
#include <hip/hip_runtime.h>
#include <hip/hip_bf16.h>
#include <math.h>

#define HIDDEN   128
#define NRBF     20
#define NLAYERS  3
#define NATOMS   50000
#define NEDGES   500000
#define CUTOFF_F 6.0f
#define PI_F     3.14159265358979323846f
#define EPB      8

typedef float v2f __attribute__((ext_vector_type(2)));
typedef float v8f __attribute__((ext_vector_type(8)));

// fast SiLU: v_exp_f32 + v_rcp_f32 + v_mul (avoids the precise-div chain)
__device__ __forceinline__ float silu_f(float x)
{
    return x * __builtin_amdgcn_rcpf(1.0f + __expf(-x));
}

// ---------------------------------------------------------------------------
// Async LDS fill (CDNA5 GLOBAL_LOAD_ASYNC_TO_LDS_B128, ASYNCcnt-tracked),
// guarded so that toolchains without the builtin fall back to a plain copy.
// Builtin signature (from hipcc diagnostic): (v4i AS1*, v4i AS3*, imm, imm)
// ---------------------------------------------------------------------------
#if defined(__gfx1250__) && __has_builtin(__builtin_amdgcn_global_load_async_to_lds_b128) && \
    __has_builtin(__builtin_amdgcn_s_wait_asynccnt)
#define USE_ASYNC_LDS 1
#else
#define USE_ASYNC_LDS 0
#endif

#if USE_ASYNC_LDS
typedef int v4i_b128 __attribute__((vector_size(16)));
typedef __attribute__((address_space(1))) v4i_b128* async_gptr_t;
typedef __attribute__((address_space(3))) v4i_b128* async_lptr_t;
#endif

__device__ __forceinline__ void lds_fill_b128(const float* g, float* l)
{
#if USE_ASYNC_LDS
    __builtin_amdgcn_global_load_async_to_lds_b128((async_gptr_t)g, (async_lptr_t)l, 0, 0);
#else
    *(float4*)l = *(const float4*)g;
#endif
}

__device__ __forceinline__ void lds_fill_wait()
{
#if USE_ASYNC_LDS
    __builtin_amdgcn_s_wait_asynccnt(0);
#endif
}

// ---------------------------------------------------------------------------
// Wave-level fp32 WMMA: C(16x16) += A(16xK, LDS, row-major lda) * B(KxNb, global,
// row-major ldb, starting at column col0).  Uses V_WMMA_F32_16X16X4_F32.
// A 16x4 layout : lane L holds M=L&15, K = k + 2*(L>>4)   (VGPR0/1 = K, K+1)
// B 4x16 layout : lane L holds N=L&15, VGPR v = row k + v + 2*(L>>4)
// C/D layout    : VGPR i = row i + 8*(L>>4), col = L&15
// ---------------------------------------------------------------------------
__device__ __forceinline__ v8f wmma_f32_tile(const float* Alds, int lda,
                                             const float* Bg, int ldb, int col0,
                                             int K, v8f acc)
{
    const int lane = threadIdx.x & 31;
    const int half = lane >> 4;
    const int l16  = lane & 15;
    const float* arow = Alds + l16 * lda;
    const float* bcol = Bg + col0 + l16;
#pragma unroll 4
    for (int k = 0; k < K; k += 4) {
        const int ka = k + 2 * half;
        v2f a, b;
        a.x = arow[ka];
        a.y = arow[ka + 1];
        b.x = bcol[ka * ldb];
        b.y = bcol[(ka + 1) * ldb];
        acc = __builtin_amdgcn_wmma_f32_16x16x4_f32(false, a, false, b,
                                                    (short)0, acc, false, false);
    }
    return acc;
}

// ---------------------------------------------------------------------------
// small elementwise kernels
// ---------------------------------------------------------------------------
__global__ void k_zero(float* p, int n)
{
    int i = blockIdx.x * blockDim.x + threadIdx.x;
    if (i < n) p[i] = 0.0f;
}

__global__ void k_copy_f(const float* src, float* dst, int n)
{
    int i = blockIdx.x * blockDim.x + threadIdx.x;
    if (i < n) dst[i] = src[i];
}

__global__ void k_copy_i2f(const int* src, float* dst, int n)
{
    int i = blockIdx.x * blockDim.x + threadIdx.x;
    if (i < n) dst[i] = (float)src[i];
}

__global__ void k_embed(const int* z, const float* table, float* ns)
{
    int i = blockIdx.x * blockDim.x + threadIdx.x;
    if (i < NATOMS * HIDDEN) {
        int n = i >> 7, h = i & 127;
        ns[i] = table[z[n] * HIDDEN + h];
    }
}

// ---------------------------------------------------------------------------
// Per-edge geometry: dist, unit vector, sinc RBF, cosine cutoff
// ---------------------------------------------------------------------------
__global__ void k_edge_geom(const float* pos, const float* cell, const int* eidx,
                            const int* coff, float* rbf, float* unitv, float* fcut,
                            float* dist_out)
{
    int e = blockIdx.x * blockDim.x + threadIdx.x;
    if (e >= NEDGES) return;
    const int row = eidx[e];
    const int col = eidx[NEDGES + e];
    const float c0 = (float)coff[e * 3 + 0];
    const float c1 = (float)coff[e * 3 + 1];
    const float c2 = (float)coff[e * 3 + 2];
    const float ox = c0 * cell[0] + c1 * cell[3] + c2 * cell[6];
    const float oy = c0 * cell[1] + c1 * cell[4] + c2 * cell[7];
    const float oz = c0 * cell[2] + c1 * cell[5] + c2 * cell[8];
    const float dx = pos[row * 3 + 0] - pos[col * 3 + 0] + ox;
    const float dy = pos[row * 3 + 1] - pos[col * 3 + 1] + oy;
    const float dz = pos[row * 3 + 2] - pos[col * 3 + 2] + oz;
    const float d   = sqrtf(dx * dx + dy * dy + dz * dz);
    const float inv = 1.0f / d;
    dist_out[e]       = d;
    unitv[e * 3 + 0]  = dx * inv;
    unitv[e * 3 + 1]  = dy * inv;
    unitv[e * 3 + 2]  = dz * inv;
    fcut[e] = (d < CUTOFF_F) ? 0.5f * (__cosf(PI_F * d / CUTOFF_F) + 1.0f) : 0.0f;
    const float w = d * (PI_F / CUTOFF_F);
#pragma unroll
    for (int n = 0; n < NRBF; n++)
        rbf[e * NRBF + n] = __sinf(w * (float)(n + 1)) * inv;
}

// ---------------------------------------------------------------------------
// Message MLP: s = silu(ns@W1 + b1) @ W2 + b2   [N,128]->[N,384]
// one block = 16 nodes, 4 waves; WMMA tiles
// ---------------------------------------------------------------------------
__global__ void __launch_bounds__(128) k_msg_mlp(const float* ns,
                                                 const float* W1, const float* b1,
                                                 const float* W2, const float* b2,
                                                 float* s)
{
    __shared__ __attribute__((aligned(16))) float A[16 * 132];
    __shared__ __attribute__((aligned(16))) float H1[16 * 132];
    const int tid  = threadIdx.x;
    const int wave = tid >> 5;
    const int lane = tid & 31;
    const int half = lane >> 4;
    const int l16  = lane & 15;
    const int node0 = blockIdx.x * 16;

    // stage A tile (16x128) into LDS with async b128 copies (16B-aligned chunks)
    for (int cid = tid; cid < 512; cid += 128) {
        const int r = cid >> 5;
        const int c = (cid & 31) * 4;
        lds_fill_b128(ns + (size_t)(node0 + r) * 128 + c, A + r * 132 + c);
    }
    lds_fill_wait();
    __syncthreads();

    for (int t = wave; t < 8; t += 4) {
        v8f acc = {};
        acc = wmma_f32_tile(A, 132, W1, 128, t * 16, 128, acc);
        const float bb = b1[t * 16 + l16];
#pragma unroll
        for (int i = 0; i < 8; i++)
            H1[(i + 8 * half) * 132 + t * 16 + l16] = silu_f(acc[i] + bb);
    }
    __syncthreads();

    for (int t = wave; t < 24; t += 4) {
        v8f acc = {};
        acc = wmma_f32_tile(H1, 132, W2, 384, t * 16, 128, acc);
        const float bb = b2[t * 16 + l16];
#pragma unroll
        for (int i = 0; i < 8; i++)
            s[(size_t)(node0 + i + 8 * half) * 384 + t * 16 + l16] = acc[i] + bb;
    }
}

// ---------------------------------------------------------------------------
// Edge message: filt = (rbf@Wf+bf)*fcut; fout = filt*s[col]; scatter-add
// 128 threads = one feature channel each; EPB edges per block
// ---------------------------------------------------------------------------
__global__ void __launch_bounds__(128) k_edge_msg(const float* rbf, const float* fcut,
                                                  const float* unitv, const int* eidx,
                                                  const float* s, const float* nv_old,
                                                  float* ns_acc, float* nv_acc,
                                                  const float* Wf, const float* bf)
{
    __shared__ __attribute__((aligned(16))) float WfL[NRBF * 384];
    __shared__ __attribute__((aligned(16))) float bfL[384];
    __shared__ float rbL[NRBF];
    __shared__ float miscL[4];   // fcut, ux, uy, uz
    __shared__ int   rcL[2];     // row, col
    const int tid = threadIdx.x;

    // stage filter weights + bias via async LDS fill
    for (int i = tid * 4; i < NRBF * 384; i += 128 * 4)
        lds_fill_b128(Wf + i, WfL + i);
    for (int i = tid * 4; i < 384; i += 128 * 4)
        lds_fill_b128(bf + i, bfL + i);
    lds_fill_wait();
    __syncthreads();

    const float bb0 = bfL[tid];
    const float bb1 = bfL[128 + tid];
    const float bb2 = bfL[256 + tid];

    const int e0 = blockIdx.x * EPB;
    for (int i = 0; i < EPB; i++) {
        const int e = e0 + i;
        __syncthreads();
        if (tid < NRBF)                 rbL[tid]        = rbf[e * NRBF + tid];
        else if (tid == 32)             miscL[0]        = fcut[e];
        else if (tid >= 33 && tid < 36) miscL[tid - 32] = unitv[e * 3 + (tid - 33)];
        else if (tid == 64)             rcL[0]          = eidx[e];
        else if (tid == 65)             rcL[1]          = eidx[NEDGES + e];
        else if (tid == 96 && e + 1 < NEDGES)
            __builtin_prefetch(&rbf[(e + 1) * NRBF], 0, 0);   // global_prefetch_b8
        __syncthreads();

        const float fc = miscL[0];
        float a0 = bb0, a1 = bb1, a2 = bb2;
#pragma unroll
        for (int k = 0; k < NRBF; k++) {
            const float r = rbL[k];
            const float* w = WfL + k * 384;
            a0 += r * w[tid];
            a1 += r * w[128 + tid];
            a2 += r * w[256 + tid];
        }
        a0 *= fc; a1 *= fc; a2 *= fc;

        const int rr = rcL[0];
        const int cc = rcL[1];
        const float* sc = s + (size_t)cc * 384;
        const float g_sv = a0 * sc[tid];
        const float g_ev = a1 * sc[128 + tid];
        const float m_s  = a2 * sc[256 + tid];

        atomicAdd(&ns_acc[(size_t)rr * 128 + tid], m_s);
        const float* nvc = nv_old + (size_t)cc * 384;
        float*       nva = nv_acc + (size_t)rr * 384;
#pragma unroll
        for (int v = 0; v < 3; v++)
            atomicAdd(&nva[v * 128 + tid], nvc[v * 128 + tid] * g_sv + g_ev * miscL[1 + v]);
    }
}

// ---------------------------------------------------------------------------
// Update part 1: Uv = nv@U, Vv = nv@V (per vector component); store Uv,
// Vnorm = ||Vv||_v, inner = sum_v Uv*Vv
// ---------------------------------------------------------------------------
__global__ void __launch_bounds__(128) k_upd_uv(const float* nv, const float* U,
                                                const float* V, float* Uv,
                                                float* Vn, float* inner)
{
    __shared__ __attribute__((aligned(16))) float A[16 * 132];
    const int tid  = threadIdx.x;
    const int wave = tid >> 5;
    const int lane = tid & 31;
    const int half = lane >> 4;
    const int l16  = lane & 15;
    const int node0 = blockIdx.x * 16;

    v8f vn2[2] = {{}, {}};
    v8f inr[2] = {{}, {}};

    for (int v = 0; v < 3; v++) {
        __syncthreads();
        for (int cid = tid; cid < 512; cid += 128) {
            const int r = cid >> 5;
            const int c = (cid & 31) * 4;
            lds_fill_b128(nv + (size_t)(node0 + r) * 384 + v * 128 + c, A + r * 132 + c);
        }
        lds_fill_wait();
        __syncthreads();
        for (int j = 0; j < 2; j++) {
            const int col0 = (wave + 4 * j) * 16;
            v8f u = {};
            u = wmma_f32_tile(A, 132, U, 128, col0, 128, u);
            v8f w = {};
            w = wmma_f32_tile(A, 132, V, 128, col0, 128, w);
#pragma unroll
            for (int i = 0; i < 8; i++)
                Uv[(size_t)(node0 + i + 8 * half) * 384 + v * 128 + col0 + l16] = u[i];
            vn2[j] += w * w;
            inr[j] += u * w;
        }
    }
    for (int j = 0; j < 2; j++) {
        const int col0 = (wave + 4 * j) * 16;
#pragma unroll
        for (int i = 0; i < 8; i++) {
            const int n = node0 + i + 8 * half;
            Vn[(size_t)n * 128 + col0 + l16]    = sqrtf(vn2[j][i]);
            inner[(size_t)n * 128 + col0 + l16] = inr[j][i];
        }
    }
}

// ---------------------------------------------------------------------------
// Update part 2: a = silu([ns,Vnorm]@W1+b1)@W2+b2; apply gates to ns, nv
// ---------------------------------------------------------------------------
__global__ void __launch_bounds__(128) k_upd_apply(float* ns, const float* Vn,
                                                   const float* inner, const float* Uv,
                                                   float* nv,
                                                   const float* W1, const float* b1,
                                                   const float* W2, const float* b2)
{
    __shared__ __attribute__((aligned(16))) float A[16 * 260];
    __shared__ __attribute__((aligned(16))) float H1[16 * 132];
    __shared__ __attribute__((aligned(16))) float Aout[16 * 388];
    const int tid  = threadIdx.x;
    const int wave = tid >> 5;
    const int lane = tid & 31;
    const int half = lane >> 4;
    const int l16  = lane & 15;
    const int node0 = blockIdx.x * 16;

    for (int cid = tid; cid < 512; cid += 128) {
        const int r = cid >> 5;
        const int c = (cid & 31) * 4;
        lds_fill_b128(ns + (size_t)(node0 + r) * 128 + c, A + r * 260 + c);
        lds_fill_b128(Vn + (size_t)(node0 + r) * 128 + c, A + r * 260 + 128 + c);
    }
    lds_fill_wait();
    __syncthreads();

    for (int t = wave; t < 8; t += 4) {
        v8f acc = {};
        acc = wmma_f32_tile(A, 260, W1, 128, t * 16, 256, acc);
        const float bb = b1[t * 16 + l16];
#pragma unroll
        for (int i = 0; i < 8; i++)
            H1[(i + 8 * half) * 132 + t * 16 + l16] = silu_f(acc[i] + bb);
    }
    __syncthreads();

    for (int t = wave; t < 24; t += 4) {
        v8f acc = {};
        acc = wmma_f32_tile(H1, 132, W2, 384, t * 16, 128, acc);
        const float bb = b2[t * 16 + l16];
#pragma unroll
        for (int i = 0; i < 8; i++)
            Aout[(i + 8 * half) * 388 + t * 16 + l16] = acc[i] + bb;
    }
    __syncthreads();

    // apply gates: thread tid owns channel h = tid for all 16 nodes
    for (int r = 0; r < 16; r++) {
        const int n = node0 + r;
        const float avv = Aout[r * 388 + tid];
        const float asv = Aout[r * 388 + 128 + tid];
        const float ass = Aout[r * 388 + 256 + tid];
        const size_t so = (size_t)n * 128 + tid;
        ns[so] += inner[so] * asv + ass;
#pragma unroll
        for (int v = 0; v < 3; v++) {
            const size_t o = (size_t)n * 384 + v * 128 + tid;
            nv[o] += avv * Uv[o];
        }
    }
}

// ---------------------------------------------------------------------------
// Head: v = silu(ns@hW1 + hb1) @ hW2 + hb2   (128->64->1, tiny: VALU)
// ---------------------------------------------------------------------------
__global__ void __launch_bounds__(64) k_head(const float* ns, const float* W1,
                                             const float* b1, const float* W2,
                                             const float* b2, float* vout)
{
    __shared__ float red[64];
    const int n = blockIdx.x;
    const int t = threadIdx.x;
    const float* x = ns + (size_t)n * 128;
    float acc = b1[t];
#pragma unroll 8
    for (int k = 0; k < 128; k++)
        acc += x[k] * W1[k * 64 + t];
    red[t] = silu_f(acc) * W2[t];
    __syncthreads();
    for (int sft = 32; sft > 0; sft >>= 1) {
        if (t < sft) red[t] += red[t + sft];
        __syncthreads();
    }
    if (t == 0) vout[n] = red[0] + b2[0];
}

// ---------------------------------------------------------------------------
extern "C" void kernel_launch(void* const* d_in, const int* in_sizes, int n_in,
                              void* d_out, int out_size, void* d_ws, size_t ws_size,
                              hipStream_t stream)
{
    (void)in_sizes; (void)n_in; (void)out_size; (void)ws_size;

    const int*   z     = (const int*)d_in[0];
    const float* pos   = (const float*)d_in[1];
    const float* cell  = (const float*)d_in[2];
    const int*   eidx  = (const int*)d_in[3];
    const int*   coff  = (const int*)d_in[4];
    const float* table = (const float*)d_in[5];
    const float* mfW   = (const float*)d_in[6];
    const float* mfb   = (const float*)d_in[7];
    const float* mW1   = (const float*)d_in[8];
    const float* mb1   = (const float*)d_in[9];
    const float* mW2   = (const float*)d_in[10];
    const float* mb2   = (const float*)d_in[11];
    const float* uU    = (const float*)d_in[12];
    const float* uV    = (const float*)d_in[13];
    const float* uW1   = (const float*)d_in[14];
    const float* ub1   = (const float*)d_in[15];
    const float* uW2   = (const float*)d_in[16];
    const float* ub2   = (const float*)d_in[17];
    const float* hW1   = (const float*)d_in[18];
    const float* hb1   = (const float*)d_in[19];
    const float* hW2   = (const float*)d_in[20];
    const float* hb2   = (const float*)d_in[21];

    // output layout: v[N] | pos[N*3] | edge_index[2E] | dist[E]
    float* out      = (float*)d_out;
    float* out_v    = out;
    float* out_pos  = out + NATOMS;
    float* out_ei   = out_pos + NATOMS * 3;
    float* out_dist = out_ei + 2 * NEDGES;

    // workspace partition (floats)
    float* ws = (float*)d_ws;
    size_t off = 0;
    float* ns    = ws + off; off += (size_t)NATOMS * 128;
    float* nvA   = ws + off; off += (size_t)NATOMS * 384;
    float* nvB   = ws + off; off += (size_t)NATOMS * 384;
    float* smsg  = ws + off; off += (size_t)NATOMS * 384;
    float* Uv    = ws + off; off += (size_t)NATOMS * 384;
    float* Vn    = ws + off; off += (size_t)NATOMS * 128;
    float* inn   = ws + off; off += (size_t)NATOMS * 128;
    float* rbf   = ws + off; off += (size_t)NEDGES * NRBF;
    float* unitv = ws + off; off += (size_t)NEDGES * 3;
    float* fcut  = ws + off; off += (size_t)NEDGES;

    const int TPB = 256;
    // init + geometry + passthrough outputs
    k_zero<<<(NATOMS * 384 + TPB - 1) / TPB, TPB, 0, stream>>>(nvA, NATOMS * 384);
    k_embed<<<(NATOMS * 128 + TPB - 1) / TPB, TPB, 0, stream>>>(z, table, ns);
    k_edge_geom<<<(NEDGES + TPB - 1) / TPB, TPB, 0, stream>>>(pos, cell, eidx, coff,
                                                              rbf, unitv, fcut, out_dist);
    k_copy_f<<<(NATOMS * 3 + TPB - 1) / TPB, TPB, 0, stream>>>(pos, out_pos, NATOMS * 3);
    k_copy_i2f<<<(2 * NEDGES + TPB - 1) / TPB, TPB, 0, stream>>>(eidx, out_ei, 2 * NEDGES);

    float* cur = nvA;
    float* alt = nvB;
    const int nodeBlocks = NATOMS / 16;  // 3125
    for (int l = 0; l < NLAYERS; l++) {
        k_msg_mlp<<<nodeBlocks, 128, 0, stream>>>(
            ns, mW1 + (size_t)l * 128 * 128, mb1 + (size_t)l * 128,
            mW2 + (size_t)l * 128 * 384, mb2 + (size_t)l * 384, smsg);

        // double-buffer nv: gather from cur, scatter-accumulate into alt
        (void)hipMemcpyAsync(alt, cur, (size_t)NATOMS * 384 * sizeof(float),
                             hipMemcpyDeviceToDevice, stream);

        k_edge_msg<<<NEDGES / EPB, 128, 0, stream>>>(
            rbf, fcut, unitv, eidx, smsg, cur, ns, alt,
            mfW + (size_t)l * NRBF * 384, mfb + (size_t)l * 384);

        k_upd_uv<<<nodeBlocks, 128, 0, stream>>>(
            alt, uU + (size_t)l * 128 * 128, uV + (size_t)l * 128 * 128, Uv, Vn, inn);

        k_upd_apply<<<nodeBlocks, 128, 0, stream>>>(
            ns, Vn, inn, Uv, alt,
            uW1 + (size_t)l * 256 * 128, ub1 + (size_t)l * 128,
            uW2 + (size_t)l * 128 * 384, ub2 + (size_t)l * 384);

        float* tmp = cur; cur = alt; alt = tmp;
    }

    k_head<<<NATOMS, 64, 0, stream>>>(ns, hW1, hb1, hW2, hb2, out_v);
}